// SMP_26998164423028
// MI455X (gfx1250) — compile-verified
//
#include <hip/hip_runtime.h>

// ---------------------------------------------------------------------------
// SMP GNN forward for MI455X (gfx1250), fp32 end-to-end.
// Bandwidth-bound problem (~2.5 GB traffic, ~14 GFLOP) -> keep f32 exactness,
// use V_WMMA_F32_16X16X4_F32 for the three big (524288-row) GEMMs per layer,
// and stage A-tiles with GLOBAL_LOAD_ASYNC_TO_LDS_B128 (ASYNCcnt).
// ---------------------------------------------------------------------------

typedef float v2f __attribute__((ext_vector_type(2)));
typedef float v8f __attribute__((ext_vector_type(8)));
typedef int v4i __attribute__((vector_size(16)));
typedef __attribute__((address_space(1))) v4i gv4i;   // global
typedef __attribute__((address_space(3))) v4i lv4i;   // LDS

constexpr int NN = 8192;            // nodes
constexpr int CC = 64;              // colors == graph size
constexpr int FH = 32;              // hidden
constexpr int EE = 65536;           // edges
constexpr int GG = 128;             // graphs
constexpr long long NC = (long long)NN * CC;   // 524288 rows
constexpr float BN_EPS = 1e-5f;

#if __has_builtin(__builtin_amdgcn_global_load_async_to_lds_b128)
#define SMP_ASYNC_LDS 1
#endif

__device__ __forceinline__ v8f wmma4(v2f a, v2f b, v8f c) {
#if __has_builtin(__builtin_amdgcn_wmma_f32_16x16x4_f32)
  // (neg_a, A, neg_b, B, c_mod, C, reuse_a, reuse_b)
  return __builtin_amdgcn_wmma_f32_16x16x4_f32(false, a, false, b, (short)0, c,
                                               false, false);
#else
  // Compile-safe fallback (gfx1250 path uses the builtin above).
  c[0] += a.x * b.x + a.y * b.y;
  return c;
#endif
}

// 16-byte global -> LDS copy; async on gfx1250 (tracked by ASYNCcnt).
__device__ __forceinline__ void copy16_g2l(const float* __restrict__ g,
                                           float* __restrict__ l) {
#ifdef SMP_ASYNC_LDS
  __builtin_amdgcn_global_load_async_to_lds_b128((gv4i*)g, (lv4i*)l, 0, 0);
#else
  *(float4*)l = *(const float4*)g;
#endif
}

__device__ __forceinline__ void wait_async_copies() {
#ifdef SMP_ASYNC_LDS
#if __has_builtin(__builtin_amdgcn_s_wait_asynccnt)
  __builtin_amdgcn_s_wait_asynccnt(0);
#else
  asm volatile("s_wait_asynccnt 0x0" ::: "memory");
#endif
#endif
}

// ------------------------------ CSR build ----------------------------------
__global__ void k_count(const int* __restrict__ dst, int* degi, int* gcnt) {
  int e = blockIdx.x * 256 + threadIdx.x;
  int d = dst[e];
  atomicAdd(&degi[d], 1);
  atomicAdd(&gcnt[d >> 6], 1);
}

__global__ void k_scan(const int* __restrict__ degi, int* offs) {
  __shared__ int tsum[1024];
  int t = threadIdx.x;
  int loc[8];
  int s = 0;
#pragma unroll
  for (int k = 0; k < 8; ++k) { loc[k] = s; s += degi[t * 8 + k]; }
  tsum[t] = s;
  __syncthreads();
  for (int o = 1; o < 1024; o <<= 1) {
    int v = (t >= o) ? tsum[t - o] : 0;
    __syncthreads();
    tsum[t] += v;
    __syncthreads();
  }
  int base = (t == 0) ? 0 : tsum[t - 1];
#pragma unroll
  for (int k = 0; k < 8; ++k) offs[t * 8 + k] = base + loc[k];
  if (t == 1023) offs[NN] = tsum[1023];
}

__global__ void k_fill(const int* __restrict__ dst, const int* __restrict__ offs,
                       int* curs, int* elist) {
  int e = blockIdx.x * 256 + threadIdx.x;
  int d = dst[e];
  int p = atomicAdd(&curs[d], 1);
  elist[offs[d] + p] = e;
}

__global__ void k_avg_edges(const int* __restrict__ gcnt, float* avgE) {
  int i = blockIdx.x * 256 + threadIdx.x;
  avgE[i] = (float)gcnt[i >> 6] * (1.0f / 64.0f);
}

// --------------------------- edge embeddings -------------------------------
// eemb = edge_attr(E,16) @ Wg(16,4) + bg
__global__ void k_edge_emb(const float* __restrict__ attr, const float* __restrict__ w,
                           const float* __restrict__ b, float* __restrict__ eemb) {
  int i = blockIdx.x * 256 + threadIdx.x;  // E*4 pairs
  int e = i >> 2, f = i & 3;
  const float* a = attr + (size_t)e * 16;
  float v = b[f];
#pragma unroll
  for (int k = 0; k < 16; ++k) v += a[k] * w[k * 4 + f];
  eemb[i] = v;
}

// emod = (eemb @ Wle(4,32) + ble) @ Wen(32,32) + ben   (per layer)
__global__ void k_edge_mod(const float* __restrict__ eemb, const float* __restrict__ lw,
                           const float* __restrict__ lb, const float* __restrict__ ew,
                           const float* __restrict__ eb, float* __restrict__ emod) {
  __shared__ float ee[8][32];
  int wv = threadIdx.x >> 5, f = threadIdx.x & 31;
  int e = blockIdx.x * 8 + wv;
  const float* a = eemb + (size_t)e * 4;
  float v = lb[f];
#pragma unroll
  for (int k = 0; k < 4; ++k) v += a[k] * lw[k * 32 + f];
  ee[wv][f] = v;
  __syncthreads();
  float o = eb[f];
#pragma unroll
  for (int k = 0; k < 32; ++k) o += ee[wv][k] * ew[k * 32 + f];
  emod[(size_t)e * 32 + f] = o;
}

// ------------------------------- init u ------------------------------------
// u[i,c,:] = b + (c==i%64) * ([1,x_i] @ W)
__global__ void k_init_u(const float* __restrict__ x, const float* __restrict__ w,
                         const float* __restrict__ b, float* __restrict__ u) {
  size_t idx = (size_t)blockIdx.x * 256 + threadIdx.x;  // NC*32
  int f = (int)(idx & 31);
  size_t row = idx >> 5;
  int node = (int)(row >> 6), c = (int)(row & 63);
  float v = b[f];
  if (c == (node & 63)) {
    v += w[f];  // constant-1 feature row 0
    const float* xr = x + (size_t)node * 15;
#pragma unroll
    for (int k = 0; k < 15; ++k) v += xr[k] * w[(k + 1) * 32 + f];
  }
  u[idx] = v;
}

// ------------------------- per-node mean / diag ----------------------------
__global__ void k_mean_diag(const float* __restrict__ u, float* __restrict__ meanb,
                            float* __restrict__ diagb) {
  int wv = threadIdx.x >> 5, lane = threadIdx.x & 31;  // one wave per node
  int node = blockIdx.x * 8 + wv;
  const float* ub = u + (size_t)node * (CC * FH);
  float s = 0.f;
  for (int c = 0; c < CC; ++c) s += ub[c * FH + lane];
  meanb[(size_t)node * FH + lane] = s * (1.0f / 64.0f);
  diagb[(size_t)node * FH + lane] = ub[(node & 63) * FH + lane];
}

// t2 = glin2(mean)+glin3(diag); t4 = glin4(diag)+glin5(mean)   (8 towers 4x4)
__global__ void k_towers(const float* __restrict__ meanb, const float* __restrict__ diagb,
                         const float* __restrict__ w2, const float* __restrict__ w3,
                         const float* __restrict__ w4, const float* __restrict__ w5,
                         float* __restrict__ t2, float* __restrict__ t4) {
  int idx = blockIdx.x * 256 + threadIdx.x;  // N*32
  int f = idx & 31, node = idx >> 5;
  int g = f >> 2, fo = f & 3;
  const float* m = meanb + (size_t)node * 32 + g * 4;
  const float* d = diagb + (size_t)node * 32 + g * 4;
  float a2 = 0, a3 = 0, a4 = 0, a5 = 0;
#pragma unroll
  for (int k = 0; k < 4; ++k) {
    int wi = g * 16 + k * 4 + fo;
    a2 += m[k] * w2[wi];
    a3 += d[k] * w3[wi];
    a4 += d[k] * w4[wi];
    a5 += m[k] * w5[wi];
  }
  t2[idx] = a2 + a3;
  t4[idx] = a4 + a5;
}

// expand (8,4,4) block-diag towers weight into dense 32x32 for WMMA
__global__ void k_expand_w1(const float* __restrict__ w1, float* __restrict__ wd) {
  int i = blockIdx.x * 256 + threadIdx.x;  // 1024
  int r = i >> 5, c = i & 31;
  wd[i] = ((r >> 2) == (c >> 2)) ? w1[(r >> 2) * 16 + (r & 3) * 4 + (c & 3)] : 0.f;
}

// ------------------------------ batch norm ---------------------------------
__global__ void k_bn_stats(const float* __restrict__ u, float* __restrict__ stats) {
  __shared__ float ss[256], sq[256];
  int t = threadIdx.x;
  size_t base = (size_t)blockIdx.x * 256 + t;
  float s = 0, q = 0;
  for (int k = 0; k < 16; ++k) {
    float v = u[base + (size_t)k * (4096ull * 256ull)];
    s += v;
    q += v * v;
  }
  ss[t] = s;
  sq[t] = q;
  __syncthreads();
  if (t < 32) {  // channel = t (stride pattern keeps channel == tid&31)
    float as = 0, aq = 0;
    for (int j = t; j < 256; j += 32) { as += ss[j]; aq += sq[j]; }
    atomicAdd(&stats[t], as);
    atomicAdd(&stats[32 + t], aq);
  }
}

__global__ void k_bn_apply(float* __restrict__ u, const float* __restrict__ stats,
                           const float* __restrict__ gamma, const float* __restrict__ beta) {
  size_t idx = (size_t)blockIdx.x * 256 + threadIdx.x;
  int f = (int)(idx & 31);
  const float invn = 1.0f / (float)NC;
  float m = stats[f] * invn;
  float var = stats[32 + f] * invn - m * m;
  float sc = rsqrtf(var + BN_EPS) * gamma[f];
  u[idx] = (u[idx] - m) * sc + beta[f];
}

// ------------------------------ WMMA GEMM ----------------------------------
// Rows = NC, Ncols = 32. MODE 0: um = u@W1d + b1 + t2[i] + (c==col(i))*t4[i]
// MODE 1: up1 = concat(um,agg)@Wup1 + b   (K=64)
// MODE 2: u   = relu(up1)@Wup2 + b + up1 + u   (residual, in place; ReLU is
//               applied at A-fragment read time so the LDS stage is a raw copy)
// LDS A-tile padded to K+4 floats/row: row stride is a 16B multiple (async
// b128 LDS writes) and stride mod 64 banks = 4 -> conflict-free column reads.
template <int K, int MODE>
__global__ __launch_bounds__(128) void k_gemm(
    const float* __restrict__ A0, const float* __restrict__ A1,
    const float* __restrict__ W, const float* __restrict__ bias,
    const float* __restrict__ t2, const float* __restrict__ t4,
    const float* __restrict__ add0, const float* __restrict__ add1,
    float* __restrict__ out) {
  __shared__ float As[64][K + 4];
  int tid = threadIdx.x;
  size_t rowbase = (size_t)blockIdx.x * 64;

  {  // stage 64xK A-tile in LDS (async global->LDS)
    int r = tid >> 1, half = tid & 1;
    size_t grow = rowbase + r;
    if (MODE == 1) {
      const float* sp = (half ? A1 : A0) + grow * 32;
#pragma unroll
      for (int j = 0; j < 32; j += 4) copy16_g2l(sp + j, &As[r][half * 32 + j]);
    } else {
      const float* sp = A0 + grow * 32 + half * 16;
#pragma unroll
      for (int j = 0; j < 16; j += 4) copy16_g2l(sp + j, &As[r][half * 16 + j]);
    }
  }

  int lane = tid & 31;
  int wv = tid >> 5;
  int r0 = wv * 16;  // wave owns 16 rows x 32 cols (two 16x16 tiles)
  constexpr int KS = K / 4;

  // preload B fragments (weights are tiny, shared by all waves)
  v2f bf[KS][2];
#pragma unroll
  for (int kk = 0; kk < KS; ++kk)
#pragma unroll
    for (int nt = 0; nt < 2; ++nt) {
      int col = (lane & 15) + nt * 16;
      int kb = kk * 4 + (lane >> 4) * 2;
      bf[kk][nt].x = W[kb * 32 + col];
      bf[kk][nt].y = W[(kb + 1) * 32 + col];
    }
  wait_async_copies();
  __syncthreads();

  v8f acc0 = {0, 0, 0, 0, 0, 0, 0, 0};
  v8f acc1 = {0, 0, 0, 0, 0, 0, 0, 0};
#pragma unroll
  for (int kk = 0; kk < KS; ++kk) {
    int kb = kk * 4 + (lane >> 4) * 2;
    int row = r0 + (lane & 15);
    v2f a;
    a.x = As[row][kb];
    a.y = As[row][kb + 1];
    if (MODE == 2) { a.x = fmaxf(a.x, 0.f); a.y = fmaxf(a.y, 0.f); }
    acc0 = wmma4(a, bf[kk][0], acc0);
    acc1 = wmma4(a, bf[kk][1], acc1);
  }

#pragma unroll
  for (int nt = 0; nt < 2; ++nt) {
    v8f acc = nt ? acc1 : acc0;
#pragma unroll
    for (int v = 0; v < 8; ++v) {
      int row = r0 + ((lane < 16) ? v : v + 8);
      size_t grow = rowbase + row;
      int col = (lane & 15) + nt * 16;
      float val = acc[v] + bias[col];
      if (MODE == 0) {
        size_t node = grow >> 6;
        int c = (int)(grow & 63);
        val += t2[node * 32 + col];
        if (c == (int)(node & 63)) val += t4[node * 32 + col];
      }
      if (MODE == 2) val += add0[grow * 32 + col] + add1[grow * 32 + col];
      out[grow * 32 + col] = val;
    }
  }
}

// --------------------- edge gather / modulate / reduce ---------------------
// agg[dst] = sum_e um[src[e]] * emod[e] / avg_edges[dst]; one block per dst.
__global__ __launch_bounds__(256) void k_aggregate(
    const float* __restrict__ um, const float* __restrict__ emod,
    const int* __restrict__ src, const int* __restrict__ offs,
    const int* __restrict__ elist, const float* __restrict__ avgE,
    float* __restrict__ agg) {
  int node = blockIdx.x;
  int t = threadIdx.x;
  float acc[8] = {0, 0, 0, 0, 0, 0, 0, 0};
  int b0 = offs[node], b1 = offs[node + 1];
  for (int ei = b0; ei < b1; ++ei) {
    int e = elist[ei];
    int s = src[e];
    const float* ub = um + (size_t)s * 2048;
    float mf = emod[(size_t)e * 32 + (t & 31)];  // (t+256k)&31 == t&31
    if (ei + 1 < b1) {
      int sn = src[elist[ei + 1]];
      __builtin_prefetch(um + (size_t)sn * 2048 + t * 8, 0, 1);
    }
#pragma unroll
    for (int k = 0; k < 8; ++k) acc[k] += ub[t + 256 * k] * mf;
  }
  float inv = 1.0f / avgE[node];
  size_t ob = (size_t)node * 2048;
#pragma unroll
  for (int k = 0; k < 8; ++k) agg[ob + t + 256 * k] = acc[k] * inv;
}

// ------------------------------ extractors ---------------------------------
// no_prop: h = ([1,x]/64)@W1 + [1,x]@W2 + b1+b2    (N,64)
__global__ void k_noprop_h(const float* __restrict__ x, const float* __restrict__ w1,
                           const float* __restrict__ b1, const float* __restrict__ w2,
                           const float* __restrict__ b2, float* __restrict__ h) {
  int i = blockIdx.x * 4 + (threadIdx.x >> 6);
  int j = threadIdx.x & 63;
  const float* xr = x + (size_t)i * 15;
  float v = b1[j] + b2[j] + (1.0f / 64.0f) * w1[j] + w2[j];
#pragma unroll
  for (int k = 1; k < 16; ++k) {
    float z = xr[k - 1];
    v += z * ((1.0f / 64.0f) * w1[k * 64 + j] + w2[k * 64 + j]);
  }
  h[(size_t)i * 64 + j] = v;
}

// feat_ext: h = mean@W1 + diag@W2 + b1+b2    (N,64)
__global__ void k_feat_h(const float* __restrict__ meanb, const float* __restrict__ diagb,
                         const float* __restrict__ w1, const float* __restrict__ b1,
                         const float* __restrict__ w2, const float* __restrict__ b2,
                         float* __restrict__ h) {
  int i = blockIdx.x * 4 + (threadIdx.x >> 6);
  int j = threadIdx.x & 63;
  const float* m = meanb + (size_t)i * 32;
  const float* d = diagb + (size_t)i * 32;
  float v = b1[j] + b2[j];
#pragma unroll
  for (int k = 0; k < 32; ++k) v += m[k] * w1[k * 64 + j] + d[k] * w2[k * 64 + j];
  h[(size_t)i * 64 + j] = v;
}

// graph mean pool + (g + relu(lin(g))), accumulate into outg
__global__ void k_pool(const float* __restrict__ h, const float* __restrict__ lw,
                       const float* __restrict__ lb, float* __restrict__ outg,
                       float scale, int accum) {
  __shared__ float sh[64];
  int g = blockIdx.x, j = threadIdx.x;
  float s = 0;
  const float* hb = h + (size_t)g * 64 * 64;
  for (int n = 0; n < 64; ++n) s += hb[n * 64 + j];
  s *= (1.0f / 64.0f);
  sh[j] = s;
  __syncthreads();
  float o = lb[j];
  for (int k = 0; k < 64; ++k) o += sh[k] * lw[k * 64 + j];
  float r = (s + fmaxf(o, 0.f)) * scale;
  if (accum) outg[g * 64 + j] += r;
  else outg[g * 64 + j] = r;
}

// head: out = relu(out@Wac + bac) + out; y = out@Wf + bf
__global__ void k_head(const float* __restrict__ outg, const float* __restrict__ acw,
                       const float* __restrict__ acb, const float* __restrict__ flw,
                       const float* __restrict__ flb, float* __restrict__ y) {
  __shared__ float sh[64], sh2[64];
  int g = blockIdx.x, j = threadIdx.x;
  sh[j] = outg[g * 64 + j];
  __syncthreads();
  float o = acb[j];
  for (int k = 0; k < 64; ++k) o += sh[k] * acw[k * 64 + j];
  float v = sh[j] + fmaxf(o, 0.f);
  sh2[j] = v * flw[j];
  __syncthreads();
  if (j == 0) {
    float s = flb[0];
    for (int k = 0; k < 64; ++k) s += sh2[k];
    y[g] = s;
  }
}

// ---------------------------------------------------------------------------
extern "C" void kernel_launch(void* const* d_in, const int* in_sizes, int n_in,
                              void* d_out, int out_size, void* d_ws, size_t ws_size,
                              hipStream_t stream) {
  (void)in_sizes; (void)n_in; (void)out_size; (void)ws_size;
  auto F = [&](int i) { return (const float*)d_in[i]; };

  const float* x = F(0);                 // (N,15)
  const int* ei = (const int*)d_in[1];   // (2,E): src then dst
  const float* edge_attr = F(2);         // (E,16)
  // d_in[3] = batch (unused; node graph = node>>6)

  // ---- params: JAX pytree flatten, sorted dict keys, None leaves dropped ----
  int p = 4;
  const float* ac_b = F(p++); const float* ac_w = F(p++);        // after_conv
  const float* fe_lin_b = F(p++); const float* fe_lin_w = F(p++);// feat_ext.lin
  const float* fe_g1_b = F(p++); const float* fe_g1_w = F(p++);  // feat_ext.uglob.lin1
  const float* fe_g2_b = F(p++); const float* fe_g2_w = F(p++);  // feat_ext.uglob.lin2
  const float* fl_b = F(p++); const float* fl_w = F(p++);        // final_lin
  const float* ge_b = F(p++); const float* ge_w = F(p++);        // global_emb
  const float* il_b = F(p++); const float* il_w = F(p++);        // initial_lin
  struct LP {
    const float *bn_beta, *bn_gamma, *en_b, *en_w, *m1b, *m1w, *m2w, *m3w,
        *m4w, *m5w, *up1b, *up1w, *up2b, *up2w, *le_b, *le_w;
  } L[4];
  for (int i = 0; i < 4; ++i) {
    L[i].bn_beta = F(p++); L[i].bn_gamma = F(p++);               // bn: beta,gamma
    L[i].en_b = F(p++); L[i].en_w = F(p++);                      // conv.edge_nn
    L[i].m1b = F(p++); L[i].m1w = F(p++);                        // conv.msg.lin1
    L[i].m2w = F(p++); L[i].m3w = F(p++);                        // lin2,lin3 (no bias)
    L[i].m4w = F(p++); L[i].m5w = F(p++);                        // lin4,lin5
    L[i].up1b = F(p++); L[i].up1w = F(p++);                      // conv.up1
    L[i].up2b = F(p++); L[i].up2w = F(p++);                      // conv.up2
    L[i].le_b = F(p++); L[i].le_w = F(p++);                      // local_emb
  }
  const float* np_lin_b = F(p++); const float* np_lin_w = F(p++);// no_prop.lin
  const float* np_g1_b = F(p++); const float* np_g1_w = F(p++);  // no_prop.uglob.lin1
  const float* np_g2_b = F(p++); const float* np_g2_w = F(p++);  // no_prop.uglob.lin2

  const int* src = ei;
  const int* dst = ei + EE;

  // ------------------------------ workspace --------------------------------
  const size_t UB = (size_t)NC * 32;  // 16,777,216 floats
  float* ws = (float*)d_ws;
  float* u = ws;
  float* um = u + UB;
  float* agg = um + UB;
  float* up1 = agg + UB;
  float* meanb = up1 + UB;                 // N*32
  float* diagb = meanb + (size_t)NN * 32;
  float* t2b = diagb + (size_t)NN * 32;
  float* t4b = t2b + (size_t)NN * 32;
  float* emod = t4b + (size_t)NN * 32;     // E*32
  float* eemb = emod + (size_t)EE * 32;    // E*4
  float* hbuf = eemb + (size_t)EE * 4;     // N*64
  float* outg = hbuf + (size_t)NN * 64;    // G*64
  float* w1d = outg + (size_t)GG * 64;     // 1024
  float* bnst = w1d + 1024;                // 64
  float* avgE = bnst + 64;                 // N
  int* degi = (int*)(avgE + NN);           // N
  int* offs = degi + NN;                   // N+1
  int* curs = offs + (NN + 1);             // N
  int* gcnt = curs + NN;                   // G
  int* elist = gcnt + GG;                  // E

  // ----------------------------- CSR + setup -------------------------------
  (void)hipMemsetAsync(degi, 0, sizeof(int) * NN, stream);
  (void)hipMemsetAsync(curs, 0, sizeof(int) * NN, stream);
  (void)hipMemsetAsync(gcnt, 0, sizeof(int) * GG, stream);
  k_count<<<EE / 256, 256, 0, stream>>>(dst, degi, gcnt);
  k_scan<<<1, 1024, 0, stream>>>(degi, offs);
  k_fill<<<EE / 256, 256, 0, stream>>>(dst, offs, curs, elist);
  k_avg_edges<<<NN / 256, 256, 0, stream>>>(gcnt, avgE);

  k_edge_emb<<<(EE * 4) / 256, 256, 0, stream>>>(edge_attr, ge_w, ge_b, eemb);

  // no_prop extractor -> outg
  k_noprop_h<<<NN / 4, 256, 0, stream>>>(x, np_g1_w, np_g1_b, np_g2_w, np_g2_b, hbuf);
  k_pool<<<GG, 64, 0, stream>>>(hbuf, np_lin_w, np_lin_b, outg, 1.0f, 0);

  // initial_lin -> u (N,64,32)
  k_init_u<<<(unsigned)(UB / 256), 256, 0, stream>>>(x, il_w, il_b, u);

  // ------------------------------- layers ----------------------------------
  for (int li = 0; li < 4; ++li) {
    const LP& lp = L[li];
    if (li > 0) {
      (void)hipMemsetAsync(bnst, 0, sizeof(float) * 64, stream);
      k_bn_stats<<<4096, 256, 0, stream>>>(u, bnst);
      k_bn_apply<<<(unsigned)(UB / 256), 256, 0, stream>>>(u, bnst, lp.bn_gamma,
                                                           lp.bn_beta);
    }
    k_edge_mod<<<EE / 8, 256, 0, stream>>>(eemb, lp.le_w, lp.le_b, lp.en_w,
                                           lp.en_b, emod);
    k_mean_diag<<<NN / 8, 256, 0, stream>>>(u, meanb, diagb);
    k_towers<<<(NN * 32) / 256, 256, 0, stream>>>(meanb, diagb, lp.m2w, lp.m3w,
                                                  lp.m4w, lp.m5w, t2b, t4b);
    k_expand_w1<<<4, 256, 0, stream>>>(lp.m1w, w1d);

    // um = utou(u): WMMA GEMM (K=32) + fused t2 / one-hot t4 epilogue
    k_gemm<32, 0><<<(unsigned)(NC / 64), 128, 0, stream>>>(
        u, nullptr, w1d, lp.m1b, t2b, t4b, nullptr, nullptr, um);

    // agg = scatter(um[src]*emod) / avg_edges
    k_aggregate<<<NN, 256, 0, stream>>>(um, emod, src, offs, elist, avgE, agg);

    // up1 = concat(um,agg) @ Wup1 + b    (K=64, WMMA)
    k_gemm<64, 1><<<(unsigned)(NC / 64), 128, 0, stream>>>(
        um, agg, lp.up1w, lp.up1b, nullptr, nullptr, nullptr, nullptr, up1);

    // u = relu(up1)@Wup2 + b + up1 + u   (K=32, WMMA, residual in place)
    k_gemm<32, 2><<<(unsigned)(NC / 64), 128, 0, stream>>>(
        up1, nullptr, lp.up2w, lp.up2b, nullptr, nullptr, up1, u, u);

    // out += extractor(u)/4
    k_mean_diag<<<NN / 8, 256, 0, stream>>>(u, meanb, diagb);
    k_feat_h<<<NN / 4, 256, 0, stream>>>(meanb, diagb, fe_g1_w, fe_g1_b, fe_g2_w,
                                         fe_g2_b, hbuf);
    k_pool<<<GG, 64, 0, stream>>>(hbuf, fe_lin_w, fe_lin_b, outg, 0.25f, 1);
  }

  // -------------------------------- head -----------------------------------
  k_head<<<GG, 64, 0, stream>>>(outg, ac_w, ac_b, fl_w, fl_b, (float*)d_out);
}